// NCSAGE_77360950935705
// MI455X (gfx1250) — compile-verified
//
#include <hip/hip_runtime.h>

typedef __attribute__((ext_vector_type(2))) float v2f;
typedef __attribute__((ext_vector_type(8))) float v8f;

#define WMMA_F32X4(a, b, c) \
  __builtin_amdgcn_wmma_f32_16x16x4_f32(false, (a), false, (b), (short)0, (c), false, false)

// ---------------------------------------------------------------- utilities
__global__ __launch_bounds__(256) void k_zero(float* __restrict__ p, size_t n) {
  size_t i = (size_t)blockIdx.x * blockDim.x + threadIdx.x;
  if (i < n) p[i] = 0.0f;
}

// ------------------------------------------------------- degree computation
// rowcnt[r] = #non-diag edges with row==r ; degScc[r] = sum cc[col] over them
__global__ __launch_bounds__(256) void k_deg_edges(const int* __restrict__ ei,
                                                   const int* __restrict__ cc,
                                                   float* __restrict__ rowcnt,
                                                   float* __restrict__ degScc,
                                                   int E) {
  int e = blockIdx.x * blockDim.x + threadIdx.x;
  if (e >= E) return;
  int c = ei[e];        // col = edge_index[0]
  int r = ei[E + e];    // row = edge_index[1]
  if (r == c) return;   // diagonal entries are replaced by the fill value
  atomicAdd(&rowcnt[r], 1.0f);
  if (cc[c]) atomicAdd(&degScc[r], 1.0f);
}

// dl  : target/cc   dh : target/rev   dll : source/cc   dhh : source/rev
__global__ __launch_bounds__(256) void k_node_d(const int* __restrict__ cc,
                                                const float* __restrict__ rowcnt,
                                                const float* __restrict__ degScc,
                                                float* __restrict__ dl,
                                                float* __restrict__ dll,
                                                float* __restrict__ dh,
                                                float* __restrict__ dhh,
                                                int n) {
  int i = blockIdx.x * blockDim.x + threadIdx.x;
  if (i >= n) return;
  float rc  = rowcnt[i];
  float sc  = degScc[i];
  float ccf = (float)cc[i];
  dl[i]  = rsqrtf(ccf * rc + 1.0f);           // deg >= 1 always (filled diag)
  dh[i]  = rsqrtf((1.0f - ccf) * rc + 1.0f);
  dll[i] = rsqrtf(sc + 1.0f);
  dhh[i] = rsqrtf((rc - sc) + 1.0f);
}

// ------------------------------------------------------------------- SpMM
// agg[i,:] = diag_i * x[i,:]   (diag = d_i^2 from the filled diagonal)
__global__ __launch_bounds__(256) void k_spmm_init(const float* __restrict__ d,
                                                   const float* __restrict__ xin,
                                                   float* __restrict__ agg,
                                                   int n) {
  size_t idx = (size_t)blockIdx.x * blockDim.x + threadIdx.x;
  if (idx >= (size_t)n * 64) return;
  size_t i = idx >> 6;
  float di = d[i];
  agg[idx] = di * di * xin[idx];
}

// agg[r,:] += d[r]*mask*d[c] * x[c,:]  ; one thread per (edge, feature)
__global__ __launch_bounds__(256) void k_spmm_edges(const int* __restrict__ ei,
                                                    const int* __restrict__ cc,
                                                    const float* __restrict__ d,
                                                    const float* __restrict__ xin,
                                                    float* __restrict__ agg,
                                                    int E, int use_row_mask, int invert) {
  size_t t = (size_t)blockIdx.x * blockDim.x + threadIdx.x;
  size_t e = t >> 6;
  int f = (int)(t & 63);
  if (e >= (size_t)E) return;
  int c = ei[e];
  int r = ei[(size_t)E + e];
  if (r == c) return;
  int m = use_row_mask ? cc[r] : cc[c];
  if (invert) m = 1 - m;
  if (!m) return;
  float val = d[r] * d[c];
  atomicAdd(&agg[(size_t)r * 64 + f], val * xin[(size_t)c * 64 + f]);
}

// ---------------------------------------------------------- WMMA f32 GEMMs
// One pass: acc(16x64) += Arows(16x64) @ W(64x64), W row-major [k][n].
__device__ __forceinline__ void gemm_pass64(const float* __restrict__ arow,
                                            const float* __restrict__ W,
                                            int l16, int ml,
                                            v8f& acc0, v8f& acc1, v8f& acc2, v8f& acc3) {
  v2f fa[16];
#pragma unroll
  for (int ks = 0; ks < 16; ++ks) fa[ks] = *(const v2f*)(arow + 4 * ks);
#pragma unroll
  for (int ks = 0; ks < 16; ++ks) {
    const int kb = 4 * ks + 2 * l16;          // this lane's K pair within slice
    const float* w = W + (size_t)kb * 64 + ml;
    v2f b0 = { w[0],  w[64 + 0]  };
    v2f b1 = { w[16], w[64 + 16] };
    v2f b2 = { w[32], w[64 + 32] };
    v2f b3 = { w[48], w[64 + 48] };
    acc0 = WMMA_F32X4(fa[ks], b0, acc0);
    acc1 = WMMA_F32X4(fa[ks], b1, acc1);
    acc2 = WMMA_F32X4(fa[ks], b2, acc2);
    acc3 = WMMA_F32X4(fa[ks], b3, acc3);
  }
}

// OUT(n_rows x 64) = A @ W1 [+ X @ W2]; optional relu. Wave computes 16 rows.
__global__ __launch_bounds__(256) void k_gemm64_dual(const float* __restrict__ A,
                                                     const float* __restrict__ W1,
                                                     const float* __restrict__ X,
                                                     const float* __restrict__ W2,
                                                     float* __restrict__ OUT,
                                                     int n_rows, int do_relu) {
  const int lane = threadIdx.x & 31;
  const int wv   = threadIdx.x >> 5;
  const int tile = blockIdx.x * 8 + wv;
  const int m0   = tile * 16;
  if (m0 >= n_rows) return;                 // uniform per wave: EXEC stays all-1s
  const int l16 = lane >> 4;
  const int ml  = lane & 15;
  int mrow = m0 + ml;
  if (mrow >= n_rows) mrow = n_rows - 1;    // clamped load, store guarded below

  v8f acc0 = {}, acc1 = {}, acc2 = {}, acc3 = {};
  gemm_pass64(A + (size_t)mrow * 64 + 2 * l16, W1, l16, ml, acc0, acc1, acc2, acc3);
  if (X) gemm_pass64(X + (size_t)mrow * 64 + 2 * l16, W2, l16, ml, acc0, acc1, acc2, acc3);

#pragma unroll
  for (int r = 0; r < 8; ++r) {
    int m = m0 + r + 8 * l16;
    if (m >= n_rows) continue;
    float v0 = acc0[r], v1 = acc1[r], v2 = acc2[r], v3 = acc3[r];
    if (do_relu) {
      v0 = fmaxf(v0, 0.0f); v1 = fmaxf(v1, 0.0f);
      v2 = fmaxf(v2, 0.0f); v3 = fmaxf(v3, 0.0f);
    }
    float* o = OUT + (size_t)m * 64 + ml;
    o[0] = v0; o[16] = v1; o[32] = v2; o[48] = v3;
  }
}

// Final classifier: OUT(n_rows x C) = A(n_rows x 64) @ W(64 x C) + bias, C<=48.
__global__ __launch_bounds__(256) void k_gemm_out(const float* __restrict__ A,
                                                  const float* __restrict__ W,
                                                  const float* __restrict__ bias,
                                                  float* __restrict__ OUT,
                                                  int n_rows, int ncols) {
  const int lane = threadIdx.x & 31;
  const int wv   = threadIdx.x >> 5;
  const int tile = blockIdx.x * 8 + wv;
  const int m0   = tile * 16;
  if (m0 >= n_rows) return;
  const int l16 = lane >> 4;
  const int ml  = lane & 15;
  int mrow = m0 + ml;
  if (mrow >= n_rows) mrow = n_rows - 1;

  const float* arow = A + (size_t)mrow * 64 + 2 * l16;
  v2f fa[16];
#pragma unroll
  for (int ks = 0; ks < 16; ++ks) fa[ks] = *(const v2f*)(arow + 4 * ks);

  v8f acc0 = {}, acc1 = {}, acc2 = {};
#pragma unroll
  for (int ks = 0; ks < 16; ++ks) {
    const int kb = 4 * ks + 2 * l16;
    int n0 = ml, n1 = 16 + ml, n2 = 32 + ml;
    v2f b0 = { W[(size_t)kb * ncols + n0], W[(size_t)(kb + 1) * ncols + n0] };
    v2f b1 = { (n1 < ncols) ? W[(size_t)kb * ncols + n1] : 0.0f,
               (n1 < ncols) ? W[(size_t)(kb + 1) * ncols + n1] : 0.0f };
    v2f b2 = { (n2 < ncols) ? W[(size_t)kb * ncols + n2] : 0.0f,
               (n2 < ncols) ? W[(size_t)(kb + 1) * ncols + n2] : 0.0f };
    acc0 = WMMA_F32X4(fa[ks], b0, acc0);
    acc1 = WMMA_F32X4(fa[ks], b1, acc1);
    acc2 = WMMA_F32X4(fa[ks], b2, acc2);
  }

#pragma unroll
  for (int r = 0; r < 8; ++r) {
    int m = m0 + r + 8 * l16;
    if (m >= n_rows) continue;
    float* o = OUT + (size_t)m * ncols;
    int n0 = ml, n1 = 16 + ml, n2 = 32 + ml;
    if (n0 < ncols) o[n0] = acc0[r] + bias[n0];
    if (n1 < ncols) o[n1] = acc1[r] + bias[n1];
    if (n2 < ncols) o[n2] = acc2[r] + bias[n2];
  }
}

// ------------------------------------------------ mixing epilogue (in place)
__global__ __launch_bounds__(256) void k_fuse(const float* __restrict__ xm,
                                              const float* __restrict__ xl,
                                              const float* __restrict__ xh,
                                              const int* __restrict__ cc,
                                              const float* __restrict__ lam1,
                                              const float* __restrict__ lam2,
                                              float* __restrict__ xf, int n) {
  size_t idx = (size_t)blockIdx.x * blockDim.x + threadIdx.x;
  if (idx >= (size_t)n * 64) return;
  size_t i = idx >> 6;
  // softmax over 2 elements, twice
  float a0 = lam1[0], a1 = lam1[1];
  float mx = fmaxf(a0, a1);
  float ea = __expf(a0 - mx), eb = __expf(a1 - mx);
  float inv = 1.0f / (ea + eb);
  float lamxl = ea * inv, laml = eb * inv;
  float b0 = lam2[0], b1 = lam2[1];
  float mx2 = fmaxf(b0, b1);
  float ec = __expf(b0 - mx2), ed = __expf(b1 - mx2);
  float inv2 = 1.0f / (ec + ed);
  float lamxh = ec * inv2, lamh = ed * inv2;
  float ccf = (float)cc[i];
  float lamx = lamxl * ccf + lamxh * (1.0f - ccf);
  xf[idx] = fmaxf(lamx * xm[idx] + laml * xl[idx] + lamh * xh[idx], 0.0f);
}

// --------------------------------------------------------------- launcher
extern "C" void kernel_launch(void* const* d_in, const int* in_sizes, int n_in,
                              void* d_out, int out_size, void* d_ws, size_t ws_size,
                              hipStream_t stream) {
  const float* x    = (const float*)d_in[0];
  const int*   ei   = (const int*)d_in[1];
  const int*   cc   = (const int*)d_in[2];
  const float* W1l  = (const float*)d_in[3];
  const float* W1r  = (const float*)d_in[4];
  const float* W2l  = (const float*)d_in[5];
  const float* W2r  = (const float*)d_in[6];
  const float* W3l  = (const float*)d_in[7];
  const float* W3r  = (const float*)d_in[8];
  const float* W4l  = (const float*)d_in[9];
  const float* W4r  = (const float*)d_in[10];
  const float* WX   = (const float*)d_in[11];
  const float* lam1 = (const float*)d_in[12];
  const float* lam2 = (const float*)d_in[13];
  const float* linW = (const float*)d_in[14];
  const float* linb = (const float*)d_in[15];

  const int N = in_sizes[0] / 64;
  const int E = in_sizes[1] / 2;
  const int C = in_sizes[15];

  float* ws     = (float*)d_ws;
  float* rowcnt = ws;
  float* degScc = ws + (size_t)N;
  float* dl     = ws + 2 * (size_t)N;
  float* dll    = ws + 3 * (size_t)N;
  float* dh     = ws + 4 * (size_t)N;
  float* dhh    = ws + 5 * (size_t)N;
  float* bufA   = ws + 6 * (size_t)N;           // N*64 agg / xm / xf scratch
  float* bufL   = bufA + (size_t)N * 64;        // low-pass branch
  float* bufH   = bufL + (size_t)N * 64;        // high-pass branch

  auto cdiv = [](size_t a, size_t b) { return (unsigned)((a + b - 1) / b); };
  dim3 b256(256);
  unsigned gElem = cdiv((size_t)N * 64, 256);
  unsigned gEdge = cdiv((size_t)E * 64, 256);
  unsigned gGemm = cdiv(((size_t)N + 15) / 16, 8);

  // degrees / normalizers (one edge pass covers all four variants)
  k_zero<<<cdiv(2 * (size_t)N, 256), b256, 0, stream>>>(rowcnt, 2 * (size_t)N);
  k_deg_edges<<<cdiv((size_t)E, 256), b256, 0, stream>>>(ei, cc, rowcnt, degScc, E);
  k_node_d<<<cdiv((size_t)N, 256), b256, 0, stream>>>(cc, rowcnt, degScc, dl, dll, dh, dhh, N);

  // low-pass branch: xl = relu(sage1(x)); xl = sage2(xl)   (in-place safe: row-local)
  k_spmm_init<<<gElem, b256, 0, stream>>>(dl, x, bufA, N);
  k_spmm_edges<<<gEdge, b256, 0, stream>>>(ei, cc, dl, x, bufA, E, 1, 0);
  k_gemm64_dual<<<gGemm, b256, 0, stream>>>(bufA, W1l, x, W1r, bufL, N, 1);
  k_spmm_init<<<gElem, b256, 0, stream>>>(dll, bufL, bufA, N);
  k_spmm_edges<<<gEdge, b256, 0, stream>>>(ei, cc, dll, bufL, bufA, E, 0, 0);
  k_gemm64_dual<<<gGemm, b256, 0, stream>>>(bufA, W2l, bufL, W2r, bufL, N, 0);

  // high-pass branch
  k_spmm_init<<<gElem, b256, 0, stream>>>(dh, x, bufA, N);
  k_spmm_edges<<<gEdge, b256, 0, stream>>>(ei, cc, dh, x, bufA, E, 1, 1);
  k_gemm64_dual<<<gGemm, b256, 0, stream>>>(bufA, W3l, x, W3r, bufH, N, 1);
  k_spmm_init<<<gElem, b256, 0, stream>>>(dhh, bufH, bufA, N);
  k_spmm_edges<<<gEdge, b256, 0, stream>>>(ei, cc, dhh, bufH, bufA, E, 0, 1);
  k_gemm64_dual<<<gGemm, b256, 0, stream>>>(bufA, W4l, bufH, W4r, bufH, N, 0);

  // xm = x @ WX  -> bufA ; xf = relu(mix) in place ; out = xf @ lin_W + b
  k_gemm64_dual<<<gGemm, b256, 0, stream>>>(x, WX, (const float*)nullptr,
                                            (const float*)nullptr, bufA, N, 0);
  k_fuse<<<gElem, b256, 0, stream>>>(bufA, bufL, bufH, cc, lam1, lam2, bufA, N);
  k_gemm_out<<<gGemm, b256, 0, stream>>>(bufA, linW, linb, (float*)d_out, N, C);
}